// De_conv_block_3D_87273735454817
// MI455X (gfx1250) — compile-verified
//
#include <hip/hip_runtime.h>

typedef __attribute__((ext_vector_type(2))) float v2f;
typedef __attribute__((ext_vector_type(8))) float v8f;

#define NEG_SLOPE 0.01f
#define TT 8
#define HH 64
#define WW 64
#define CSTRIDE (TT * HH * WW) /* 32768 spatial positions */
#define KVOL 27
#define NTILES (TT * HH * (WW / 16)) /* 2048 wave-tiles */

__device__ __forceinline__ v8f wmma_f32(v2f a, v2f b, v8f c) {
  // V_WMMA_F32_16X16X4_F32 : D(16x16 f32) = A(16x4 f32) * B(4x16 f32) + C
  return __builtin_amdgcn_wmma_f32_16x16x4_f32(false, a, false, b, (short)0, c,
                                               false, false);
}

// ---------------------------------------------------------------------------
// Repack activation (64, 32768) -> channel-pair interleaved (32, 32768, 2)
// so a WMMA B-fragment (two adjacent channels, one spatial point) is one b64.
// ---------------------------------------------------------------------------
__global__ __launch_bounds__(256) void repack_pairs(
    const float* __restrict__ src, float* __restrict__ dst) {
  const int idx = blockIdx.x * 256 + threadIdx.x; // pair*CSTRIDE + sp
  if (idx >= 32 * CSTRIDE) return;
  const int pair = idx / CSTRIDE;
  const int sp = idx - pair * CSTRIDE;
  v2f v;
  v.x = src[(size_t)(2 * pair) * CSTRIDE + sp];
  v.y = src[(size_t)(2 * pair + 1) * CSTRIDE + sp];
  ((v2f*)dst)[idx] = v;
}

// ---------------------------------------------------------------------------
// Repack weights (OC, 64, K) -> (K, 32, 64, 2), zero-padded to 64 out rows,
// so a WMMA A-fragment is one b64 and no OC guard is needed in the GEMM loop.
// ---------------------------------------------------------------------------
__global__ __launch_bounds__(256) void repack_weight(
    const float* __restrict__ src, float* __restrict__ dst, int OC, int K) {
  const int idx = blockIdx.x * 256 + threadIdx.x; // (k*32+pair)*64 + o
  if (idx >= K * 32 * 64) return;
  const int o = idx & 63;
  const int pair = (idx >> 6) & 31;
  const int k = idx >> 11;
  const int c = pair * 2;
  v2f v;
  v.x = (o < OC) ? src[((size_t)o * 64 + c) * K + k] : 0.f;
  v.y = (o < OC) ? src[((size_t)o * 64 + c + 1) * K + k] : 0.f;
  ((v2f*)dst)[idx] = v;
}

// ---------------------------------------------------------------------------
// Dense 3x3x3 conv, pad=1, Cin=64 -> OC, optional lrelu.
// In/out activations and weights are pair-interleaved. One wave computes a
// 64(Cout) x 16(w) tile with 4 f32-WMMA accumulators. Boundary handling is
// clamp-address + multiply-by-mask: all loads are unconditional.
// ---------------------------------------------------------------------------
__global__ __launch_bounds__(256) void conv3d_k3_wmma(
    const float* __restrict__ xp,   // (32, 32768, 2) interleaved
    const float* __restrict__ wp,   // (27, 32, 64, 2) padded
    const float* __restrict__ bias, // (OC)
    float* __restrict__ yout,       // (32, 32768, 2) interleaved
    int OC, int do_lrelu) {
  const int lane = threadIdx.x & 31;
  const int wave = threadIdx.x >> 5;
  const int tile = blockIdx.x * 8 + wave;
  if (tile >= NTILES) return;

  const int w0 = (tile & 3) << 4;
  const int h = (tile >> 2) & (HH - 1);
  const int t = tile >> 8;
  const int n = lane & 15;         // B column / A row within 16-tile
  const int kp = lane >> 4;        // K-pair select (0/1) per f32 WMMA layout

  const v2f* xv = (const v2f*)xp;
  const v2f* wv = (const v2f*)wp;

  v8f acc0 = {}, acc1 = {}, acc2 = {}, acc3 = {};
  const int wi_n = w0 + n;

  for (int k = 0; k < KVOL; ++k) {
    const int kt = k / 9, kh = (k / 3) % 3, kw = k % 3;
    const int ti = t + kt - 1;
    const int hi = h + kh - 1;
    const int wi = wi_n + kw - 1;
    const bool ok =
        (ti >= 0) & (ti < TT) & (hi >= 0) & (hi < HH) & (wi >= 0) & (wi < WW);
    const float fmask = ok ? 1.f : 0.f;
    const int tic = ti < 0 ? 0 : (ti > TT - 1 ? TT - 1 : ti);
    const int hic = hi < 0 ? 0 : (hi > HH - 1 ? HH - 1 : hi);
    const int wic = wi < 0 ? 0 : (wi > WW - 1 ? WW - 1 : wi);
    const int spc = (tic * HH + hic) * WW + wic;
    const int kbase = k * 32;

#pragma unroll 4
    for (int cb = 0; cb < 64; cb += 4) {
      const int pb = (cb >> 1) + kp; // channel-pair index = cA>>1
      v2f b = xv[(size_t)pb * CSTRIDE + spc];
      b.x *= fmask;
      b.y *= fmask;
      const v2f* ap = wv + (size_t)(kbase + pb) * 64 + n;
      v2f a0 = ap[0];
      v2f a1 = ap[16];
      v2f a2 = ap[32];
      v2f a3 = ap[48];
      acc0 = wmma_f32(a0, b, acc0);
      acc1 = wmma_f32(a1, b, acc1);
      acc2 = wmma_f32(a2, b, acc2);
      acc3 = wmma_f32(a3, b, acc3);
    }
  }

  // D layout: VGPR j, lanes 0-15 -> M=j, lanes 16-31 -> M=j+8; N = lane%16
  const int rbase = kp * 8;
  const int spo = t * HH * WW + h * WW + w0 + n;
#define STORE_CONV(mt)                                                    \
  for (int j = 0; j < 8; ++j) {                                           \
    const int o = (mt)*16 + rbase + j;                                    \
    if (o < OC) {                                                         \
      float v = acc##mt[j] + bias[o];                                     \
      if (do_lrelu) v = (v >= 0.f) ? v : NEG_SLOPE * v;                   \
      yout[((size_t)(o >> 1) * CSTRIDE + spo) * 2 + (o & 1)] = v;         \
    }                                                                     \
  }
  STORE_CONV(0) STORE_CONV(1) STORE_CONV(2) STORE_CONV(3)
#undef STORE_CONV
}

// ---------------------------------------------------------------------------
// Deformable 3x3x3 conv (h/w offsets, bilinear) fused with bias+lrelu and the
// 1x1x1 residual conv on x:  out = lrelu(deform(y1)+bd) + (Wr*x + br)
// y1/offsets/x and all weights pair-interleaved; output standard layout.
// ---------------------------------------------------------------------------
__global__ __launch_bounds__(256) void deform_res_wmma(
    const float* __restrict__ y1p,  // (32, 32768, 2)
    const float* __restrict__ offp, // (27, 32768, 2): (.x=dh, .y=dw)
    const float* __restrict__ wdp,  // (27, 32, 64, 2)
    const float* __restrict__ bd,   // (64)
    const float* __restrict__ xp,   // (32, 32768, 2)
    const float* __restrict__ wrp,  // (32, 64, 2)
    const float* __restrict__ br,   // (64)
    float* __restrict__ out) {      // (64, T, H, W) standard
  const int lane = threadIdx.x & 31;
  const int wave = threadIdx.x >> 5;
  const int tile = blockIdx.x * 8 + wave;
  if (tile >= NTILES) return;

  const int w0 = (tile & 3) << 4;
  const int h = (tile >> 2) & (HH - 1);
  const int t = tile >> 8;
  const int n = lane & 15;
  const int kp = lane >> 4;

  const v2f* yv = (const v2f*)y1p;
  const v2f* ov = (const v2f*)offp;
  const v2f* wdv = (const v2f*)wdp;
  const v2f* xv = (const v2f*)xp;
  const v2f* wrv = (const v2f*)wrp;

  v8f acc0 = {}, acc1 = {}, acc2 = {}, acc3 = {};
  const int wcol = w0 + n;
  const int spo = t * HH * WW + h * WW + wcol;

  for (int k = 0; k < KVOL; ++k) {
    const int kt = k / 9, kh = (k / 3) % 3, kw = k % 3;
    const int ti = t + kt - 1;
    const bool tok = (ti >= 0) & (ti < TT);
    const int tc = ti < 0 ? 0 : (ti > TT - 1 ? TT - 1 : ti);

    const v2f dv = ov[(size_t)k * CSTRIDE + spo]; // (dh, dw) channel pair
    const float hs = (float)(h + kh - 1) + dv.x;
    const float ws = (float)(wcol + kw - 1) + dv.y;
    const float hf = floorf(hs), wf = floorf(ws);
    const float fh = hs - hf, fw = ws - wf;
    const int h0 = (int)hf, wq = (int)wf;
    const int h1 = h0 + 1, w1 = wq + 1;
    const bool h0ok = (h0 >= 0) & (h0 < HH), h1ok = (h1 >= 0) & (h1 < HH);
    const bool w0ok = (wq >= 0) & (wq < WW), w1ok = (w1 >= 0) & (w1 < WW);
    const int h0c = h0 < 0 ? 0 : (h0 > HH - 1 ? HH - 1 : h0);
    const int h1c = h1 < 0 ? 0 : (h1 > HH - 1 ? HH - 1 : h1);
    const int w0c = wq < 0 ? 0 : (wq > WW - 1 ? WW - 1 : wq);
    const int w1c = w1 < 0 ? 0 : (w1 > WW - 1 ? WW - 1 : w1);
    const int base = tc * HH * WW;
    const int i00 = base + h0c * WW + w0c;
    const int i01 = base + h0c * WW + w1c;
    const int i10 = base + h1c * WW + w0c;
    const int i11 = base + h1c * WW + w1c;
    const float g00 = (tok & h0ok & w0ok) ? (1.f - fh) * (1.f - fw) : 0.f;
    const float g01 = (tok & h0ok & w1ok) ? (1.f - fh) * fw : 0.f;
    const float g10 = (tok & h1ok & w0ok) ? fh * (1.f - fw) : 0.f;
    const float g11 = (tok & h1ok & w1ok) ? fh * fw : 0.f;
    const int kbase = k * 32;

#pragma unroll 2
    for (int cb = 0; cb < 64; cb += 4) {
      const int pb = (cb >> 1) + kp;
      const v2f* yc = yv + (size_t)pb * CSTRIDE;
      const v2f t00 = yc[i00];
      const v2f t01 = yc[i01];
      const v2f t10 = yc[i10];
      const v2f t11 = yc[i11];
      v2f b;
      b.x = g00 * t00.x + g01 * t01.x + g10 * t10.x + g11 * t11.x;
      b.y = g00 * t00.y + g01 * t01.y + g10 * t10.y + g11 * t11.y;
      const v2f* ap = wdv + (size_t)(kbase + pb) * 64 + n;
      v2f a0 = ap[0];
      v2f a1 = ap[16];
      v2f a2 = ap[32];
      v2f a3 = ap[48];
      acc0 = wmma_f32(a0, b, acc0);
      acc1 = wmma_f32(a1, b, acc1);
      acc2 = wmma_f32(a2, b, acc2);
      acc3 = wmma_f32(a3, b, acc3);
    }
  }

  // bd + leaky-relu in the accumulators, then WMMA-accumulate the residual
  const int rbase = kp * 8;
#define LRELU_ACC(mt)                                                    \
  for (int j = 0; j < 8; ++j) {                                          \
    const int o = (mt)*16 + rbase + j;                                   \
    float v = acc##mt[j] + bd[o];                                        \
    acc##mt[j] = (v >= 0.f) ? v : NEG_SLOPE * v;                         \
  }
  LRELU_ACC(0) LRELU_ACC(1) LRELU_ACC(2) LRELU_ACC(3)
#undef LRELU_ACC

  // residual 1x1x1 conv on x: K = 64 channels
#pragma unroll 4
  for (int cb = 0; cb < 64; cb += 4) {
    const int pb = (cb >> 1) + kp;
    v2f b = xv[(size_t)pb * CSTRIDE + spo];
    const v2f* ap = wrv + (size_t)pb * 64 + n;
    v2f a0 = ap[0];
    v2f a1 = ap[16];
    v2f a2 = ap[32];
    v2f a3 = ap[48];
    acc0 = wmma_f32(a0, b, acc0);
    acc1 = wmma_f32(a1, b, acc1);
    acc2 = wmma_f32(a2, b, acc2);
    acc3 = wmma_f32(a3, b, acc3);
  }

#define STORE_OUT(mt)                                                    \
  for (int j = 0; j < 8; ++j) {                                          \
    const int o = (mt)*16 + rbase + j;                                   \
    out[(size_t)o * CSTRIDE + spo] = acc##mt[j] + br[o];                 \
  }
  STORE_OUT(0) STORE_OUT(1) STORE_OUT(2) STORE_OUT(3)
#undef STORE_OUT
}

// ---------------------------------------------------------------------------
extern "C" void kernel_launch(void* const* d_in, const int* in_sizes, int n_in,
                              void* d_out, int out_size, void* d_ws,
                              size_t ws_size, hipStream_t stream) {
  const float* x = (const float*)d_in[0];
  const float* W1 = (const float*)d_in[1];
  const float* b1 = (const float*)d_in[2];
  const float* W_off = (const float*)d_in[3];
  const float* b_off = (const float*)d_in[4];
  const float* Wd = (const float*)d_in[5];
  const float* bd = (const float*)d_in[6];
  const float* Wr = (const float*)d_in[7];
  const float* br = (const float*)d_in[8];
  float* out = (float*)d_out;

  // workspace carve-up (floats)
  float* p = (float*)d_ws;
  float* xp = p;   p += (size_t)64 * CSTRIDE;      // 8 MB
  float* y1p = p;  p += (size_t)64 * CSTRIDE;      // 8 MB
  float* offp = p; p += (size_t)54 * CSTRIDE;      // 6.75 MB
  float* w1p = p;  p += (size_t)KVOL * 32 * 64 * 2; // 442 KB
  float* wofp = p; p += (size_t)KVOL * 32 * 64 * 2; // 442 KB
  float* wdp = p;  p += (size_t)KVOL * 32 * 64 * 2; // 442 KB
  float* wrp = p;  p += (size_t)32 * 64 * 2;        // 16 KB

  const dim3 blk(256);

  // one-time repacks (pair-interleave activations, fragment-order weights)
  repack_pairs<<<dim3((32 * CSTRIDE + 255) / 256), blk, 0, stream>>>(x, xp);
  repack_weight<<<dim3((KVOL * 32 * 64 + 255) / 256), blk, 0, stream>>>(
      W1, w1p, 64, KVOL);
  repack_weight<<<dim3((KVOL * 32 * 64 + 255) / 256), blk, 0, stream>>>(
      W_off, wofp, 54, KVOL);
  repack_weight<<<dim3((KVOL * 32 * 64 + 255) / 256), blk, 0, stream>>>(
      Wd, wdp, 64, KVOL);
  repack_weight<<<dim3((1 * 32 * 64 + 255) / 256), blk, 0, stream>>>(
      Wr, wrp, 64, 1);

  const dim3 grd(NTILES / 8); // 256 blocks, 8 waves each, 1 tile per wave

  // Stage 1: y1 = lrelu(conv3x3x3(x, W1) + b1)         (interleaved out)
  conv3d_k3_wmma<<<grd, blk, 0, stream>>>(xp, w1p, b1, y1p, 64, 1);
  // Stage 2: offsets = conv3x3x3(y1, W_off) + b_off    (interleaved out)
  conv3d_k3_wmma<<<grd, blk, 0, stream>>>(y1p, wofp, b_off, offp, 54, 0);
  // Stage 3: out = lrelu(deform(y1, offs, Wd) + bd) + (Wr*x + br)
  deform_res_wmma<<<grd, blk, 0, stream>>>(y1p, offp, wdp, bd, xp, wrp, br,
                                           out);
}